// SpMM_CPU_16338055594697
// MI455X (gfx1250) — compile-verified
//
#include <hip/hip_runtime.h>

// ---------------------------------------------------------------------------
// COO SpMM for MI455X (gfx1250):  out[row[e]] += values[e] * x[col[e]], d=32.
// Memory-bound (AI ~0.23 FLOP/B): x and out are L2-resident (12.8 MB each vs
// 192 MB L2). Strategy: lane==feature (wave32 == d), async-copy edge tiles
// into LDS (double buffered, ASYNCcnt), b128 LDS broadcasts of edge metadata,
// 8-deep batched 128B-coalesced x gathers, global_atomic_add_f32 scatter.
// ---------------------------------------------------------------------------

#ifndef __has_builtin
#define __has_builtin(x) 0
#endif

#if __has_builtin(__builtin_amdgcn_global_load_async_to_lds_b128) && \
    __has_builtin(__builtin_amdgcn_s_wait_asynccnt)
#define USE_ASYNC_LDS 1
#else
#define USE_ASYNC_LDS 0
#endif

#define TILE   2048               // edges staged per block per iteration
#define BLOCK  256                // 8 waves (wave32)
#define EPW    (TILE / (BLOCK / 32))   // 256 edges per wave per tile

// clang ext-vector v4i32 pointers in the address spaces the builtin expects
typedef int v4i __attribute__((ext_vector_type(4)));
typedef v4i __attribute__((address_space(1)))* gv4_t;   // global (AS1)
typedef v4i __attribute__((address_space(3)))* lv4_t;   // LDS (AS3)

__global__ void __launch_bounds__(BLOCK) spmm_coo_kernel(
    const int*   __restrict__ row,
    const int*   __restrict__ col,
    const float* __restrict__ val,
    const float* __restrict__ x,
    float*       __restrict__ out,
    int E, int numTiles)
{
  __shared__ int   s_row[2][TILE];
  __shared__ int   s_col[2][TILE];
  __shared__ float s_val[2][TILE];

  const int tid  = threadIdx.x;
  const int lane = tid & 31;
  const int wv   = tid >> 5;

  const int lastV4 = (E >> 2) - 1;   // last valid int4 index into edge arrays

  // Stage one tile of edge data (row/col/val) into LDS buffer b.
  auto stage = [&](int b, int t) {
#pragma unroll
    for (int k = 0; k < (TILE >> 2) / BLOCK; ++k) {       // 2 iterations
      int i4 = tid + k * BLOCK;                           // int4 slot within tile
      int g4 = t * (TILE >> 2) + i4;                      // global int4 index
      if (g4 > lastV4) g4 = lastV4;   // clamp: stays in-bounds; clamped slots
                                      // are never read (tail uses global loads)
#if USE_ASYNC_LDS
      __builtin_amdgcn_global_load_async_to_lds_b128(
          (gv4_t)row + g4, (lv4_t)&s_row[b][i4 << 2], 0, 0);
      __builtin_amdgcn_global_load_async_to_lds_b128(
          (gv4_t)col + g4, (lv4_t)&s_col[b][i4 << 2], 0, 0);
      __builtin_amdgcn_global_load_async_to_lds_b128(
          (gv4_t)val + g4, (lv4_t)&s_val[b][i4 << 2], 0, 0);
#else
      int4   r4 = ((const int4*)row)[g4];
      int4   c4 = ((const int4*)col)[g4];
      float4 v4 = ((const float4*)val)[g4];
      *(int4*)  &s_row[b][i4 << 2] = r4;
      *(int4*)  &s_col[b][i4 << 2] = c4;
      *(float4*)&s_val[b][i4 << 2] = v4;
#endif
    }
  };

  int t = blockIdx.x;
  if (t >= numTiles) return;

  int cur = 0;
  stage(cur, t);                       // prologue: fill buffer 0

  while (true) {
#if USE_ASYNC_LDS
    __builtin_amdgcn_s_wait_asynccnt(0);   // my async copies for buf[cur] done
#endif
    __syncthreads();                   // all waves' copies done; prev tile fully read

    int tn = t + gridDim.x;
    if (tn < numTiles) stage(cur ^ 1, tn); // overlap next tile's DMA with compute

    // ---- compute tile t from LDS buffer cur: wave wv owns edges [wv*EPW, +EPW)
    const int ebase = t * TILE + wv * EPW;
    if (ebase + EPW <= E) {
      // Full tile (common case): batch 8 edges — 6 ds_load_b128 broadcasts,
      // 8 independent x gathers in flight, 8 fire-and-forget f32 atomics.
      for (int j = 0; j < EPW; j += 8) {
        const int idx = wv * EPW + j;
        int4   ra = *(const int4*)  &s_row[cur][idx];      // ds_load_b128
        int4   rb = *(const int4*)  &s_row[cur][idx + 4];
        int4   ca = *(const int4*)  &s_col[cur][idx];
        int4   cb = *(const int4*)  &s_col[cur][idx + 4];
        float4 va = *(const float4*)&s_val[cur][idx];
        float4 vb = *(const float4*)&s_val[cur][idx + 4];
        float x0 = x[(ca.x << 5) + lane];                  // clause of 8 loads
        float x1 = x[(ca.y << 5) + lane];
        float x2 = x[(ca.z << 5) + lane];
        float x3 = x[(ca.w << 5) + lane];
        float x4 = x[(cb.x << 5) + lane];
        float x5 = x[(cb.y << 5) + lane];
        float x6 = x[(cb.z << 5) + lane];
        float x7 = x[(cb.w << 5) + lane];
        unsafeAtomicAdd(&out[(ra.x << 5) + lane], va.x * x0);
        unsafeAtomicAdd(&out[(ra.y << 5) + lane], va.y * x1);
        unsafeAtomicAdd(&out[(ra.z << 5) + lane], va.z * x2);
        unsafeAtomicAdd(&out[(ra.w << 5) + lane], va.w * x3);
        unsafeAtomicAdd(&out[(rb.x << 5) + lane], vb.x * x4);
        unsafeAtomicAdd(&out[(rb.y << 5) + lane], vb.y * x5);
        unsafeAtomicAdd(&out[(rb.z << 5) + lane], vb.z * x6);
        unsafeAtomicAdd(&out[(rb.w << 5) + lane], vb.w * x7);
      }
    } else {                           // tail tile — wave-uniform global loads
      for (int j = 0; j < EPW; ++j) {
        int e = ebase + j;
        if (e < E) {
          int   r  = row[e];           // uniform -> s_load via constant cache
          int   c  = col[e];
          float v  = val[e];
          float xv = x[(c << 5) + lane];
          unsafeAtomicAdd(&out[(r << 5) + lane], v * xv);
        }
      }
    }

    if (tn >= numTiles) break;
    cur ^= 1;
    t = tn;
  }
}

__global__ void __launch_bounds__(BLOCK) zero_f32_kernel(float* __restrict__ out, int n) {
  int i4 = (blockIdx.x * blockDim.x + threadIdx.x) * 4;
  if (i4 + 3 < n) {
    *(float4*)(out + i4) = make_float4(0.f, 0.f, 0.f, 0.f);
  } else {
    for (int k = i4; k < n; ++k) out[k] = 0.f;
  }
}

extern "C" void kernel_launch(void* const* d_in, const int* in_sizes, int n_in,
                              void* d_out, int out_size, void* d_ws, size_t ws_size,
                              hipStream_t stream) {
  const int*   row = (const int*)  d_in[0];
  const int*   col = (const int*)  d_in[1];
  const float* val = (const float*)d_in[2];
  const float* x   = (const float*)d_in[3];
  float*       out = (float*)d_out;
  const int E = in_sizes[0];

  // Harness poisons d_out; atomically-accumulated output must start at zero.
  {
    int nthreads = (out_size + 3) / 4;
    int zb = (nthreads + BLOCK - 1) / BLOCK;
    if (zb < 1) zb = 1;
    zero_f32_kernel<<<zb, BLOCK, 0, stream>>>(out, out_size);
  }

  const int numTiles = (E + TILE - 1) / TILE;
  int grid = numTiles < 1024 ? numTiles : 1024;
  if (grid < 1) grid = 1;
  spmm_coo_kernel<<<grid, BLOCK, 0, stream>>>(row, col, val, x, out, E, numTiles);
}